// REGCN_35450660061272
// MI455X (gfx1250) — compile-verified
//
#include <hip/hip_runtime.h>

// ---------------------------------------------------------------------------
// RE-GCN forward for MI455X (gfx1250).  All GEMMs use V_WMMA_F32_16X16X4_F32
// to stay on the CDNA5 matrix pipe while preserving fp32 numerics.
// GEMM: 128x64 block tile, 8 waves (4x2), each wave a 32x32 quadrant
// (4 f32 accumulators) -> 4 WMMAs per 8 LDS dwords in the K inner loop.
// ---------------------------------------------------------------------------

typedef float v2f __attribute__((ext_vector_type(2)));
typedef float v8f __attribute__((ext_vector_type(8)));

#define HDIM   256
#define NENT   20000
#define NRELS  200
#define EDGES  50000
#define LHIST  3
#define TQ     2048
#define CCH    50
#define NLAYER 2
#define CHUNK  512

__device__ __forceinline__ float sigmf(float x) { return 1.0f / (1.0f + expf(-x)); }

// ---------------------------------------------------------------------------
// WMMA f32 GEMM:  D[M x N] = A[M x K] @ W[N x K]^T (+bias[N] if bias != null)
// Requires K % 16 == 0 and lda/ldw multiples of 4 (float4 global loads).
// ---------------------------------------------------------------------------
__global__ __launch_bounds__(256)
void k_gemm_wmma_f32(const float* __restrict__ A, int lda,
                     const float* __restrict__ W, int ldw,
                     float* __restrict__ D, long long ldd,
                     int M, int N, int K,
                     const float* __restrict__ bias)
{
    __shared__ float As[128][17];
    __shared__ float Bs[64][17];

    const int tid  = threadIdx.x;
    const int wave = tid >> 5;
    const int lane = tid & 31;
    const int wm   = wave >> 1;        // 0..3 : 32-row band
    const int wn   = wave & 1;         // 0..1 : 32-col band
    const int bm   = blockIdx.y * 128;
    const int bn   = blockIdx.x * 64;
    const int lrow = lane & 15;
    const int lhi  = lane >> 4;        // 0 or 1

    v8f acc00 = {};  // rows wm*32..+15, cols wn*32..+15
    v8f acc01 = {};  // rows wm*32..+15, cols wn*32+16..+31
    v8f acc10 = {};  // rows wm*32+16..+31, cols wn*32..+15
    v8f acc11 = {};  // rows wm*32+16..+31, cols wn*32+16..+31

    // A loader: 128 rows x 16 cols, 8 floats/thread
    const int ra = tid >> 1, ca = (tid & 1) * 8;
    // B loader: 64 rows x 16 cols, 4 floats/thread
    const int rb = tid >> 2, cb = (tid & 3) * 4;

    for (int kb = 0; kb < K; kb += 16) {
        {
            float4 av0 = make_float4(0.f, 0.f, 0.f, 0.f);
            float4 av1 = make_float4(0.f, 0.f, 0.f, 0.f);
            const int ga = bm + ra;
            if (ga < M) {
                const float* ap = A + (size_t)ga * lda + kb + ca;
                av0 = *(const float4*)(ap);
                av1 = *(const float4*)(ap + 4);
            }
            As[ra][ca + 0] = av0.x; As[ra][ca + 1] = av0.y;
            As[ra][ca + 2] = av0.z; As[ra][ca + 3] = av0.w;
            As[ra][ca + 4] = av1.x; As[ra][ca + 5] = av1.y;
            As[ra][ca + 6] = av1.z; As[ra][ca + 7] = av1.w;

            float4 bv = make_float4(0.f, 0.f, 0.f, 0.f);
            const int gb = bn + rb;
            if (gb < N) bv = *(const float4*)(W + (size_t)gb * ldw + kb + cb);
            Bs[rb][cb + 0] = bv.x; Bs[rb][cb + 1] = bv.y;
            Bs[rb][cb + 2] = bv.z; Bs[rb][cb + 3] = bv.w;
        }
        __syncthreads();

        #pragma unroll
        for (int ks = 0; ks < 16; ks += 4) {
            const int kk = ks + 2 * lhi;
            v2f a0, a1, b0, b1;
            // A 16x4 fragment: a[j] = A[lrow][kk+j]
            a0.x = As[wm * 32 + lrow][kk + 0];
            a0.y = As[wm * 32 + lrow][kk + 1];
            a1.x = As[wm * 32 + 16 + lrow][kk + 0];
            a1.y = As[wm * 32 + 16 + lrow][kk + 1];
            // B 4x16 fragment (B = W^T): b[j] = W[col][kk+j]
            b0.x = Bs[wn * 32 + lrow][kk + 0];
            b0.y = Bs[wn * 32 + lrow][kk + 1];
            b1.x = Bs[wn * 32 + 16 + lrow][kk + 0];
            b1.y = Bs[wn * 32 + 16 + lrow][kk + 1];
            acc00 = __builtin_amdgcn_wmma_f32_16x16x4_f32(
                        false, a0, false, b0, (short)0, acc00, false, false);
            acc01 = __builtin_amdgcn_wmma_f32_16x16x4_f32(
                        false, a0, false, b1, (short)0, acc01, false, false);
            acc10 = __builtin_amdgcn_wmma_f32_16x16x4_f32(
                        false, a1, false, b0, (short)0, acc10, false, false);
            acc11 = __builtin_amdgcn_wmma_f32_16x16x4_f32(
                        false, a1, false, b1, (short)0, acc11, false, false);
        }
        __syncthreads();
    }

    const int row0 = bm + wm * 32;
    const int col0 = bn + wn * 32 + lrow;
    const int col1 = col0 + 16;
    float bias0 = 0.f, bias1 = 0.f;
    if (bias) {
        if (col0 < N) bias0 = bias[col0];
        if (col1 < N) bias1 = bias[col1];
    }
    #pragma unroll
    for (int v = 0; v < 8; ++v) {
        const int r0 = row0 + v + 8 * lhi;
        const int r1 = r0 + 16;
        if (r0 < M) {
            if (col0 < N) D[(size_t)r0 * ldd + col0] = acc00[v] + bias0;
            if (col1 < N) D[(size_t)r0 * ldd + col1] = acc01[v] + bias1;
        }
        if (r1 < M) {
            if (col0 < N) D[(size_t)r1 * ldd + col0] = acc10[v] + bias0;
            if (col1 < N) D[(size_t)r1 * ldd + col1] = acc11[v] + bias1;
        }
    }
}

static inline void gemm(const float* A, int lda, const float* W, int ldw,
                        float* D, long long ldd, int M, int N, int K,
                        const float* bias, hipStream_t s)
{
    dim3 g((N + 63) / 64, (M + 127) / 128);
    k_gemm_wmma_f32<<<g, 256, 0, s>>>(A, lda, W, ldw, D, ldd, M, N, K, bias);
}

// ---------------------------------------------------------------------------
// Elementwise / reduction kernels
// ---------------------------------------------------------------------------

// row-wise L2 normalize (rows x 256), safe in-place
__global__ void k_l2norm(const float* __restrict__ x, float* __restrict__ y)
{
    const int r = blockIdx.x, h = threadIdx.x;
    const float v = x[(size_t)r * HDIM + h];
    __shared__ float red[256];
    red[h] = v * v;
    __syncthreads();
    for (int s = 128; s > 0; s >>= 1) {
        if (h < s) red[h] += red[h + s];
        __syncthreads();
    }
    const float inv = 1.0f / fmaxf(sqrtf(red[0]), 1e-12f);
    y[(size_t)r * HDIM + h] = v * inv;
}

// set relation-incidence mask (idempotent plain stores -> deterministic)
__global__ void k_mask_set(const int* __restrict__ src, const int* __restrict__ dst,
                           const int* __restrict__ rid, unsigned char* __restrict__ mask)
{
    const int e = blockIdx.x * blockDim.x + threadIdx.x;
    if (e >= EDGES) return;
    const int r = rid[e];
    mask[(size_t)r * NENT + src[e]] = 1;
    mask[(size_t)r * NENT + dst[e]] = 1;
}

// mean of features of nodes incident to each relation (set semantics)
__global__ void k_rel_agg(const unsigned char* __restrict__ mask,
                          const float* __restrict__ ent,
                          float* __restrict__ rel_ent)
{
    const int r = blockIdx.x, h = threadIdx.x;
    const unsigned char* mrow = mask + (size_t)r * NENT;
    float sum = 0.f;
    int cnt = 0;
    for (int n = 0; n < NENT; ++n) {
        if (mrow[n]) { sum += ent[(size_t)n * HDIM + h]; ++cnt; }
    }
    rel_ent[(size_t)r * HDIM + h] = (cnt > 0) ? (sum / (float)cnt) : 0.0f;
}

// gru_in[r] = concat(rel_ent[r], raw rel_embeds[r])
__global__ void k_build_gruin(const float* __restrict__ rel_ent,
                              const float* __restrict__ rel_raw,
                              float* __restrict__ gru_in)
{
    const int r = blockIdx.x, h = threadIdx.x;
    gru_in[(size_t)r * 512 + h]       = rel_ent[(size_t)r * HDIM + h];
    gru_in[(size_t)r * 512 + 256 + h] = rel_raw[(size_t)r * HDIM + h];
}

__global__ void k_gru_combine(const float* __restrict__ gi, const float* __restrict__ gh,
                              const float* __restrict__ hprev, float* __restrict__ out)
{
    const int r = blockIdx.x, h = threadIdx.x;
    const float ir = gi[(size_t)r * 768 + h];
    const float iz = gi[(size_t)r * 768 + 256 + h];
    const float in = gi[(size_t)r * 768 + 512 + h];
    const float hr = gh[(size_t)r * 768 + h];
    const float hz = gh[(size_t)r * 768 + 256 + h];
    const float hn = gh[(size_t)r * 768 + 512 + h];
    const float rg = sigmf(ir + hr);
    const float z  = sigmf(iz + hz);
    const float n  = tanhf(in + rg * hn);
    const float hp = hprev[(size_t)r * HDIM + h];
    out[(size_t)r * HDIM + h] = (1.0f - z) * n + z * hp;
}

// scatter-add (nf[src] + n_rel[rid]) into pre[dst]; deg[dst] += 1
__global__ void k_scatter(const int* __restrict__ src, const int* __restrict__ dst,
                          const int* __restrict__ rid,
                          const float* __restrict__ nf, const float* __restrict__ nrel,
                          float* __restrict__ pre, float* __restrict__ deg)
{
    const int e = blockIdx.x, h = threadIdx.x;
    const int s = src[e], d = dst[e], r = rid[e];
    const float v = nf[(size_t)s * HDIM + h] + nrel[(size_t)r * HDIM + h];
    atomicAdd(&pre[(size_t)d * HDIM + h], v);
    if (h == 0) atomicAdd(&deg[d], 1.0f);
}

__global__ void k_premean(float* __restrict__ pre, const float* __restrict__ deg)
{
    const int n = blockIdx.x, h = threadIdx.x;
    pre[(size_t)n * HDIM + h] /= fmaxf(deg[n], 1.0f);
}

// h = Y1 + (deg==0 ? Y3 : Y2); eval-mode rrelu
__global__ void k_rgcn_combine(const float* __restrict__ Y1, const float* __restrict__ Y2,
                               const float* __restrict__ Y3, const float* __restrict__ deg,
                               float* __restrict__ out)
{
    const int n = blockIdx.x, h = threadIdx.x;
    const size_t i = (size_t)n * HDIM + h;
    const float s = (deg[n] == 0.0f) ? Y3[i] : Y2[i];
    const float v = Y1[i] + s;
    const float slope = ((1.0f / 8.0f) + (1.0f / 3.0f)) * 0.5f;
    out[i] = (v >= 0.0f) ? v : slope * v;
}

// ent = ent + sigmoid(Upre) * (nf - ent)
__global__ void k_ent_update(float* __restrict__ ent, const float* __restrict__ Upre,
                             const float* __restrict__ nf)
{
    const int n = blockIdx.x, h = threadIdx.x;
    const size_t i = (size_t)n * HDIM + h;
    const float u = sigmf(Upre[i]);
    const float e = ent[i];
    ent[i] = e + u * (nf[i] - e);
}

// decoder: gather [ent[ia[t]], B[ib[t]]] and apply bn0 affine
__global__ void k_gather_pair(const float* __restrict__ ent, const float* __restrict__ B,
                              const int* __restrict__ ia, const int* __restrict__ ib,
                              const float* __restrict__ g0, const float* __restrict__ b0,
                              float* __restrict__ xc, int t0)
{
    const int i = blockIdx.x, h = threadIdx.x;
    const int t = t0 + i;
    const float bnrs = rsqrtf(1.0f + 1e-5f);
    const float x0 = ent[(size_t)ia[t] * HDIM + h];
    const float x1 = B[(size_t)ib[t] * HDIM + h];
    xc[(size_t)i * 512 + h]       = x0 * (g0[0] * bnrs) + b0[0];
    xc[(size_t)i * 512 + 256 + h] = x1 * (g0[1] * bnrs) + b0[1];
}

// conv1d 2->50 ch, k=3, pad=1, + convb, + bn1 affine + relu -> fm[i][c*256+h]
__global__ void k_conv(const float* __restrict__ xc, const float* __restrict__ convW,
                       const float* __restrict__ convb,
                       const float* __restrict__ g1, const float* __restrict__ b1,
                       float* __restrict__ fm)
{
    const int i = blockIdx.x, h = threadIdx.x;
    __shared__ float xs[2][258];
    if (h == 0) { xs[0][0] = 0.f; xs[0][257] = 0.f; xs[1][0] = 0.f; xs[1][257] = 0.f; }
    xs[0][h + 1] = xc[(size_t)i * 512 + h];
    xs[1][h + 1] = xc[(size_t)i * 512 + 256 + h];
    __syncthreads();
    const float bnrs = rsqrtf(1.0f + 1e-5f);
    for (int c = 0; c < CCH; ++c) {
        float acc = convb[c];
        #pragma unroll
        for (int ci = 0; ci < 2; ++ci)
            #pragma unroll
            for (int k = 0; k < 3; ++k)
                acc += xs[ci][h + k] * convW[(c * 2 + ci) * 3 + k];
        acc = acc * (g1[c] * bnrs) + b1[c];
        fm[(size_t)i * (CCH * HDIM) + c * HDIM + h] = fmaxf(acc, 0.0f);
    }
}

// hid = relu(hid * g2 + b2)   (per-column affine)
__global__ void k_bnrelu(float* __restrict__ hid, const float* __restrict__ g2,
                         const float* __restrict__ b2)
{
    const int i = blockIdx.x, h = threadIdx.x;
    const float bnrs = rsqrtf(1.0f + 1e-5f);
    const size_t idx = (size_t)i * HDIM + h;
    hid[idx] = fmaxf(hid[idx] * (g2[h] * bnrs) + b2[h], 0.0f);
}

// ---------------------------------------------------------------------------
// Host orchestration
// ---------------------------------------------------------------------------
struct CTParams {
    const float *g0, *b0, *convW, *convb, *g1, *b1, *linW, *linb, *g2, *b2;
};

extern "C" void kernel_launch(void* const* d_in, const int* in_sizes, int n_in,
                              void* d_out, int out_size, void* d_ws, size_t ws_size,
                              hipStream_t stream)
{
    (void)in_sizes; (void)n_in; (void)out_size; (void)ws_size;

    // --- inputs (setup_inputs dict insertion order; ct dicts leaf-flattened) ---
    const float* ent_embeds = (const float*)d_in[0];
    const float* rel_embeds = (const float*)d_in[1];
    const float* rgcn_W1    = (const float*)d_in[2];   // (2,256,256)
    const float* rgcn_W2    = (const float*)d_in[3];
    const float* rgcn_W3    = (const float*)d_in[4];
    const float* lin_W      = (const float*)d_in[5];
    const float* lin_b      = (const float*)d_in[6];
    const float* gru_Wih    = (const float*)d_in[7];   // (768,512)
    const float* gru_Whh    = (const float*)d_in[8];   // (768,256)
    const float* gru_bih    = (const float*)d_in[9];
    const float* gru_bhh    = (const float*)d_in[10];
    CTParams obj_ct = { (const float*)d_in[11], (const float*)d_in[12],
                        (const float*)d_in[13], (const float*)d_in[14],
                        (const float*)d_in[15], (const float*)d_in[16],
                        (const float*)d_in[17], (const float*)d_in[18],
                        (const float*)d_in[19], (const float*)d_in[20] };
    CTParams rel_ct = { (const float*)d_in[21], (const float*)d_in[22],
                        (const float*)d_in[23], (const float*)d_in[24],
                        (const float*)d_in[25], (const float*)d_in[26],
                        (const float*)d_in[27], (const float*)d_in[28],
                        (const float*)d_in[29], (const float*)d_in[30] };
    const int* src_all = (const int*)d_in[31];
    const int* dst_all = (const int*)d_in[32];
    const int* rid_all = (const int*)d_in[33];
    const int* subj    = (const int*)d_in[34];
    const int* relq    = (const int*)d_in[35];
    const int* objq    = (const int*)d_in[36];
    float* outf = (float*)d_out;

    // --- workspace bump allocator (256B aligned) ---
    char* wsp = (char*)d_ws;
    auto alloc = [&](size_t bytes) -> char* {
        char* p = wsp;
        wsp += (bytes + 255) & ~(size_t)255;
        return p;
    };
    const size_t EB = (size_t)NENT * HDIM * sizeof(float);
    float* ent   = (float*)alloc(EB);
    float* nfa   = (float*)alloc(EB);
    float* nfb   = (float*)alloc(EB);
    float* pre   = (float*)alloc(EB);
    float* Y1    = (float*)alloc(EB);
    float* Y2    = (float*)alloc(EB);
    float* Y3    = (float*)alloc(EB);
    float* deg   = (float*)alloc((size_t)NENT * sizeof(float));
    unsigned char* mask = (unsigned char*)alloc((size_t)NRELS * NENT);
    float* rel_e  = (float*)alloc((size_t)NRELS * HDIM * sizeof(float));
    float* rel_en = (float*)alloc((size_t)NRELS * HDIM * sizeof(float));
    float* n_rel  = (float*)alloc((size_t)NRELS * HDIM * sizeof(float));
    float* gru_in = (float*)alloc((size_t)NRELS * 512 * sizeof(float));
    float* gi     = (float*)alloc((size_t)NRELS * 768 * sizeof(float));
    float* gh     = (float*)alloc((size_t)NRELS * 768 * sizeof(float));
    float* xc     = (float*)alloc((size_t)CHUNK * 512 * sizeof(float));
    float* fm     = (float*)alloc((size_t)CHUNK * CCH * HDIM * sizeof(float));
    float* hid    = (float*)alloc((size_t)CHUNK * HDIM * sizeof(float));

    // --- initial normalization ---
    k_l2norm<<<NENT, 256, 0, stream>>>(ent_embeds, ent);
    k_l2norm<<<NRELS, 256, 0, stream>>>(rel_embeds, rel_e);

    // --- history snapshots ---
    for (int t = 0; t < LHIST; ++t) {
        const int* st = src_all + (size_t)t * EDGES;
        const int* dt = dst_all + (size_t)t * EDGES;
        const int* rt = rid_all + (size_t)t * EDGES;

        // relation aggregation (set semantics via byte mask)
        hipMemsetAsync(mask, 0, (size_t)NRELS * NENT, stream);
        k_mask_set<<<(EDGES + 255) / 256, 256, 0, stream>>>(st, dt, rt, mask);
        k_rel_agg<<<NRELS, 256, 0, stream>>>(mask, ent, rel_en);

        // GRU over relations
        k_build_gruin<<<NRELS, 256, 0, stream>>>(rel_en, rel_embeds, gru_in);
        gemm(gru_in, 512, gru_Wih, 512, gi, 768, NRELS, 768, 512, gru_bih, stream);
        gemm(rel_e, 256, gru_Whh, 256, gh, 768, NRELS, 768, 256, gru_bhh, stream);
        k_gru_combine<<<NRELS, 256, 0, stream>>>(gi, gh, rel_e, n_rel);
        k_l2norm<<<NRELS, 256, 0, stream>>>(n_rel, n_rel);

        // RGCN layers (segment-mean commuted before the W1 GEMM)
        hipMemcpyAsync(nfa, ent, EB, hipMemcpyDeviceToDevice, stream);
        float* nf_in = nfa;
        float* nf_out = nfb;
        for (int l = 0; l < NLAYER; ++l) {
            hipMemsetAsync(pre, 0, EB, stream);
            hipMemsetAsync(deg, 0, (size_t)NENT * sizeof(float), stream);
            k_scatter<<<EDGES, 256, 0, stream>>>(st, dt, rt, nf_in, n_rel, pre, deg);
            k_premean<<<NENT, 256, 0, stream>>>(pre, deg);
            const float* W1l = rgcn_W1 + (size_t)l * HDIM * HDIM;
            const float* W2l = rgcn_W2 + (size_t)l * HDIM * HDIM;
            const float* W3l = rgcn_W3 + (size_t)l * HDIM * HDIM;
            gemm(pre,   256, W1l, 256, Y1, 256, NENT, 256, 256, nullptr, stream);
            gemm(nf_in, 256, W2l, 256, Y2, 256, NENT, 256, 256, nullptr, stream);
            gemm(nf_in, 256, W3l, 256, Y3, 256, NENT, 256, 256, nullptr, stream);
            k_rgcn_combine<<<NENT, 256, 0, stream>>>(Y1, Y2, Y3, deg, nf_out);
            float* tmp = nf_in; nf_in = nf_out; nf_out = tmp;
        }
        k_l2norm<<<NENT, 256, 0, stream>>>(nf_in, nf_in);

        // gated entity update
        gemm(ent, 256, lin_W, 256, Y1, 256, NENT, 256, 256, lin_b, stream);
        k_ent_update<<<NENT, 256, 0, stream>>>(ent, Y1, nf_in);

        hipMemcpyAsync(rel_e, n_rel, (size_t)NRELS * HDIM * sizeof(float),
                       hipMemcpyDeviceToDevice, stream);
    }

    // --- ConvTransE decoders ---
    for (int dec = 0; dec < 2; ++dec) {
        const CTParams& ct = dec ? rel_ct : obj_ct;
        const float* Bsecond = dec ? ent : rel_e;      // pair[1] source
        const int* ib        = dec ? objq : relq;
        const float* Wlogit  = dec ? rel_e : ent;
        const int Nlog       = dec ? NRELS : NENT;
        float* outbase       = dec ? (outf + (size_t)TQ * NENT) : outf;
        for (int c0 = 0; c0 < TQ; c0 += CHUNK) {
            k_gather_pair<<<CHUNK, 256, 0, stream>>>(ent, Bsecond, subj, ib,
                                                     ct.g0, ct.b0, xc, c0);
            k_conv<<<CHUNK, 256, 0, stream>>>(xc, ct.convW, ct.convb, ct.g1, ct.b1, fm);
            gemm(fm, CCH * HDIM, ct.linW, CCH * HDIM, hid, 256,
                 CHUNK, 256, CCH * HDIM, ct.linb, stream);
            k_bnrelu<<<CHUNK, 256, 0, stream>>>(hid, ct.g2, ct.b2);
            gemm(hid, 256, Wlogit, 256, outbase + (size_t)c0 * Nlog, Nlog,
                 CHUNK, Nlog, 256, nullptr, stream);
        }
    }
}